// RetrievalHeadPyramidBottomUpInstantSimple_14894946583374
// MI455X (gfx1250) — compile-verified
//
#include <hip/hip_runtime.h>
#include <hip/hip_bf16.h>

// ---------------------------------------------------------------------------
// Math: mean(bilinear_upsample(f)) == mean(f) exactly for half-pixel triangle
// resize with integer scale (column sums of the interpolation matrix are
// uniform).  So the reference collapses to per-channel spatial means followed
// by a [32,3840] x [3840,1024] f32 GEMM.
//
// Stage 1 (bandwidth bound, ~193 MB @ 23.3 TB/s): per-(b,c) plane mean,
//   coalesced b128 non-temporal loads (streamed once; keep L2 for conv_w),
//   wave32 shuffle reduce + tiny LDS cross-wave reduce.
// Stage 2 (compute trivially small): V_WMMA_F32_16X16X4_F32, one wave per
//   16x16 output tile, K advanced 4 at a time, full f32 accumulation.
// ---------------------------------------------------------------------------

typedef __attribute__((ext_vector_type(2))) float v2f;
typedef __attribute__((ext_vector_type(4))) float v4f;   // native clang vector (OK for nontemporal builtin)
typedef __attribute__((ext_vector_type(8))) float v8f;

// ---------------------------- Stage 1: pooling -----------------------------
// One block per (b, c) spatial plane.  VEC4 path uses 16B loads (all levels
// with HW % 4 == 0 are also 16B-aligned per plane); level 3 (HW=49) scalar.
template <int BLOCK, bool VEC4>
__global__ __launch_bounds__(BLOCK)
void pool_mean_kernel(const float* __restrict__ src, float* __restrict__ v,
                      int C, int HW, float inv_hw, int coff) {
  __shared__ float red[BLOCK / 32];
  const int bc = blockIdx.x;
  const int b = bc / C;
  const int c = bc - b * C;
  const float* base = src + (size_t)bc * (size_t)HW;

  float s = 0.0f;
  if (VEC4) {
    const int n4 = HW >> 2;
    const v4f* p = (const v4f*)base;
    for (int i = threadIdx.x; i < n4; i += BLOCK) {
      v4f x = __builtin_nontemporal_load(p + i);   // TH=NT: streamed once
      s += (x.x + x.y) + (x.z + x.w);
    }
  } else {
    for (int i = threadIdx.x; i < HW; i += BLOCK) {
      s += __builtin_nontemporal_load(base + i);
    }
  }

  // wave32 butterfly reduce
  #pragma unroll
  for (int off = 16; off > 0; off >>= 1) s += __shfl_down(s, off, 32);

  const int lane = threadIdx.x & 31;
  const int wave = threadIdx.x >> 5;
  if (lane == 0) red[wave] = s;
  __syncthreads();
  if (threadIdx.x == 0) {
    float t = 0.0f;
    #pragma unroll
    for (int w = 0; w < BLOCK / 32; ++w) t += red[w];
    v[(size_t)b * 3840 + coff + c] = t * inv_hw;
  }
}

// ----------------------- Stage 2: WMMA f32 GEMM ----------------------------
// out[32,1024] = v[32,3840] * conv_w[1024,3840]^T
// One wave per 16x16 output tile: 2 M-tiles x 64 N-tiles = 128 waves.
//
// V_WMMA_F32_16X16X4_F32 fragment layout (ISA 7.12.2, wave32):
//   A (16x4):  lane l, vgpr r:  M = l%16, K = 2*(l/16) + r
//   B (4x16):  lane l, vgpr r:  N = l%16, K = 2*(l/16) + r   (mirror of A)
//   C/D:       lane l, vgpr r:  N = l%16, M = r + 8*(l/16)
__global__ __launch_bounds__(32)
void wmma_gemm_kernel(const float* __restrict__ v,      // [32 x 3840]
                      const float* __restrict__ w,      // [1024 x 3840] row-major
                      float* __restrict__ out) {        // [32 x 1024]
  const int tile  = blockIdx.x;      // 0..127
  const int mtile = tile & 1;        // 0..1
  const int ntile = tile >> 1;       // 0..63
  const int lane  = threadIdx.x;     // 0..31 (EXEC all ones: block == 1 wave)
  const int half  = lane >> 4;
  const int l16   = lane & 15;

  // Per-lane base pointers: contiguous float2 per K-step (8B aligned).
  const float* arow = v + (size_t)(mtile * 16 + l16) * 3840 + 2 * half;
  const float* brow = w + (size_t)(ntile * 16 + l16) * 3840 + 2 * half;

  v8f acc = {};
  #pragma unroll 8
  for (int k = 0; k < 3840; k += 4) {
    v2f a = *(const v2f*)(arow + k);
    v2f b = *(const v2f*)(brow + k);
    // 8 args: (neg_a, A, neg_b, B, c_mod, C, reuse_a, reuse_b)
    acc = __builtin_amdgcn_wmma_f32_16x16x4_f32(false, a, false, b,
                                                (short)0, acc, false, false);
  }

  const int ncol = ntile * 16 + l16;
  #pragma unroll
  for (int r = 0; r < 8; ++r) {
    const int m = mtile * 16 + r + 8 * half;
    out[(size_t)m * 1024 + ncol] = acc[r];
  }
}

// ------------------------------ launcher -----------------------------------
extern "C" void kernel_launch(void* const* d_in, const int* in_sizes, int n_in,
                              void* d_out, int out_size, void* d_ws, size_t ws_size,
                              hipStream_t stream) {
  const float* feat0  = (const float*)d_in[0];   // [32, 256, 56, 56]
  const float* feat1  = (const float*)d_in[1];   // [32, 512, 28, 28]
  const float* feat2  = (const float*)d_in[2];   // [32,1024, 14, 14]
  const float* feat3  = (const float*)d_in[3];   // [32,2048,  7,  7]
  const float* conv_w = (const float*)d_in[4];   // [1024, 3840]
  float* out = (float*)d_out;                    // [32, 1024]
  float* v   = (float*)d_ws;                     // [32, 3840] scratch (491,520 B)

  const int B = 32;
  // Level 0: C=256, HW=3136
  pool_mean_kernel<256, true ><<<B * 256,  256, 0, stream>>>(feat0, v, 256,  3136, 1.0f / 3136.0f, 0);
  // Level 1: C=512, HW=784
  pool_mean_kernel<256, true ><<<B * 512,  256, 0, stream>>>(feat1, v, 512,   784, 1.0f / 784.0f,  256);
  // Level 2: C=1024, HW=196
  pool_mean_kernel<128, true ><<<B * 1024, 128, 0, stream>>>(feat2, v, 1024,  196, 1.0f / 196.0f,  768);
  // Level 3: C=2048, HW=49 (not /4 -> scalar loads)
  pool_mean_kernel< 64, false><<<B * 2048,  64, 0, stream>>>(feat3, v, 2048,   49, 1.0f / 49.0f,  1792);

  // GEMM: 128 tiles, one wave32 each.
  wmma_gemm_kernel<<<128, 32, 0, stream>>>(v, conv_w, out);
}